// LocalNet_15247133901599
// MI455X (gfx1250) — compile-verified
//
#include <hip/hip_runtime.h>

// ---------------------------------------------------------------------------
// Binary-tree pairwise reduction, depth 16, per-pair Linear(2,1) + ReLU.
// Bandwidth-bound (512 MB streamed once, ~0.5 GFLOP) -> optimize for the
// 23.3 TB/s HBM path: async global->LDS streaming pipeline (CDNA5
// GLOBAL_LOAD_ASYNC_TO_LDS + s_wait_asynccnt), conflict-free LDS layout,
// wave32 VALU fma/relu. WMMA is intentionally not used: per-pair weights mean
// there is no shared B matrix, so matrix ops cannot express this contraction.
// ---------------------------------------------------------------------------

#define TREE_DEPTH 16
#define BATCH      2048
#define COLS       (1 << TREE_DEPTH)     // 65536
#define TILE_COLS  4096
#define TILES      (COLS / TILE_COLS)    // 16
#define NBUF       3                     // triple buffer
#define BLOCK      256
#define CH         8                     // async b64 instructions / thread / tile

struct WPtrs { const float* w[TREE_DEPTH]; };

// ---- CDNA5 async global->LDS copy (8 bytes per lane) ----------------------
typedef int v2i __attribute__((ext_vector_type(2)));
typedef __attribute__((address_space(1))) v2i* g_v2i_ptr;
typedef __attribute__((address_space(3))) v2i* l_v2i_ptr;

__device__ __forceinline__ void async_b64(const float* g, float* l) {
#if __has_builtin(__builtin_amdgcn_global_load_async_to_lds_b64)
  __builtin_amdgcn_global_load_async_to_lds_b64(
      (g_v2i_ptr)g, (l_v2i_ptr)l, 0, 0);
#else
  unsigned lo = (unsigned)(unsigned long long)
      (__attribute__((address_space(3))) void*)l;
  asm volatile("global_load_async_to_lds_b64 %0, %1, off"
               :: "v"(lo), "v"(g) : "memory");
#endif
}

#if __has_builtin(__builtin_amdgcn_s_wait_asynccnt)
#define WAIT_ASYNC(n) __builtin_amdgcn_s_wait_asynccnt(n)
#else
#define WAIT_ASYNC(n) asm volatile("s_wait_asynccnt %0" :: "n"(n) : "memory")
#endif

__device__ __forceinline__ float relu_fma(float a, float wa, float b, float wb) {
  return fmaxf(fmaf(b, wb, a * wa), 0.0f);
}

__global__ __launch_bounds__(BLOCK)
void tree_reduce_kernel(const float* __restrict__ x, WPtrs wp,
                        float* __restrict__ out) {
  // Pair-major tile layout: pair (thread t, slot j) lives at dword j*512+2t.
  // ds_load_b64 per slot j then touches banks {2t, 2t+1} -> all 64 banks, no
  // conflicts. Async writes scatter into this layout per-lane.
  __shared__ __align__(16) float xbuf[NBUF][TILE_COLS];     // 3 x 16 KB
  __shared__ __align__(16) float vals[TILES * BLOCK];       // 4096 level-4 inputs
  __shared__ __align__(16) float pong[TILES * BLOCK / 2];   // 2048 ping-pong

  const int   tid  = threadIdx.x;
  const int   row  = blockIdx.x;
  const float* rowp = x + (size_t)row * COLS;

  // Per-lane LDS destination base (dwords) for async scatter:
  // global pair P = q*256 + tid  ->  owner T = P/8, slot j = P%8
  // dest dword = j*512 + 2T = (tid&7)*512 + (tid>>3)*2 + 64*q
  const int dstBase = (tid & 7) * 512 + (tid >> 3) * 2;

  auto issueTile = [&](int k) {
    const float* src = rowp + (size_t)k * TILE_COLS + tid * 2;
    float*       dst = &xbuf[k % NBUF][dstBase];
#pragma unroll
    for (int q = 0; q < CH; ++q)
      async_b64(src + q * 512, dst + q * 64);   // 256B contiguous per wave/instr
  };

  issueTile(0);
  issueTile(1);

  // ---- Phase A: stream tiles, fold levels 0..3 in registers ---------------
  for (int k = 0; k < TILES; ++k) {
    if (k < TILES - 1) { WAIT_ASYNC(CH); }   // oldest tile retired, 1 in flight
    else               { WAIT_ASYNC(0);  }
    __syncthreads();                          // tile k visible to all waves

    const float2* xb = (const float2*)&xbuf[k % NBUF][0];

    // Weights for this thread's 16 columns (base col = k*4096 + 16*tid).
    const float* w0 = wp.w[0] + k * TILE_COLS       + (tid << 4); // 16 floats
    const float* w1 = wp.w[1] + k * (TILE_COLS / 2) + (tid << 3); //  8 floats
    const float* w2 = wp.w[2] + k * (TILE_COLS / 4) + (tid << 2); //  4 floats
    const float* w3 = wp.w[3] + k * (TILE_COLS / 8) + (tid << 1); //  2 floats
    float4 wa0 = ((const float4*)w0)[0];
    float4 wa1 = ((const float4*)w0)[1];
    float4 wa2 = ((const float4*)w0)[2];
    float4 wa3 = ((const float4*)w0)[3];
    float4 ub0 = ((const float4*)w1)[0];
    float4 ub1 = ((const float4*)w1)[1];
    float4 uc  = *(const float4*)w2;
    float2 ud  = *(const float2*)w3;

    float2 p0 = xb[0 * 256 + tid], p1 = xb[1 * 256 + tid];
    float2 p2 = xb[2 * 256 + tid], p3 = xb[3 * 256 + tid];
    float2 p4 = xb[4 * 256 + tid], p5 = xb[5 * 256 + tid];
    float2 p6 = xb[6 * 256 + tid], p7 = xb[7 * 256 + tid];

    // level 0
    float q0 = relu_fma(p0.x, wa0.x, p0.y, wa0.y);
    float q1 = relu_fma(p1.x, wa0.z, p1.y, wa0.w);
    float q2 = relu_fma(p2.x, wa1.x, p2.y, wa1.y);
    float q3 = relu_fma(p3.x, wa1.z, p3.y, wa1.w);
    float q4 = relu_fma(p4.x, wa2.x, p4.y, wa2.y);
    float q5 = relu_fma(p5.x, wa2.z, p5.y, wa2.w);
    float q6 = relu_fma(p6.x, wa3.x, p6.y, wa3.y);
    float q7 = relu_fma(p7.x, wa3.z, p7.y, wa3.w);
    // level 1
    float r0 = relu_fma(q0, ub0.x, q1, ub0.y);
    float r1 = relu_fma(q2, ub0.z, q3, ub0.w);
    float r2 = relu_fma(q4, ub1.x, q5, ub1.y);
    float r3 = relu_fma(q6, ub1.z, q7, ub1.w);
    // level 2
    float s0 = relu_fma(r0, uc.x, r1, uc.y);
    float s1 = relu_fma(r2, uc.z, r3, uc.w);
    // level 3
    vals[k * BLOCK + tid] = relu_fma(s0, ud.x, s1, ud.y);

    __syncthreads();   // all reads of xbuf[k%NBUF] done -> safe to refill below
    if (k + NBUF - 1 < TILES) issueTile(k + NBUF - 1);
  }

  // ---- Phase B: levels 4..15 across the row (4096 -> 1) -------------------
  float* bin  = vals;
  float* bout = pong;
  int    m    = TILES * BLOCK;   // 4096 inputs at level 4
#pragma unroll 1
  for (int lvl = 4; lvl < TREE_DEPTH; ++lvl) {
    const int     nout = m >> 1;
    const float2* in2  = (const float2*)bin;                 // banks 2j,2j+1: conflict-free
    const float2* wl   = (const float2*)wp.w[lvl];           // global pair j -> flats 2j,2j+1
    for (int j = tid; j < nout; j += BLOCK) {
      float2 a = in2[j];
      float2 w = wl[j];
      float  u = fmaf(a.y, w.y, a.x * w.x);
      bout[j]  = (lvl < TREE_DEPTH - 1) ? fmaxf(u, 0.0f) : u;  // no ReLU on last level
    }
    __syncthreads();
    float* t = bin; bin = bout; bout = t;
    m = nout;
  }

  if (tid == 0) out[row] = bin[0];
}

extern "C" void kernel_launch(void* const* d_in, const int* in_sizes, int n_in,
                              void* d_out, int out_size, void* d_ws, size_t ws_size,
                              hipStream_t stream) {
  (void)in_sizes; (void)n_in; (void)out_size; (void)d_ws; (void)ws_size;
  const float* x = (const float*)d_in[0];
  WPtrs wp;
  for (int i = 0; i < TREE_DEPTH; ++i) wp.w[i] = (const float*)d_in[1 + i];
  float* out = (float*)d_out;

  tree_reduce_kernel<<<dim3(BATCH), dim3(BLOCK), 0, stream>>>(x, wp, out);
}